// SpatialAttentionModule_46084999086084
// MI455X (gfx1250) — compile-verified
//
#include <hip/hip_runtime.h>

typedef __attribute__((ext_vector_type(2))) float v2f;
typedef __attribute__((ext_vector_type(8))) float v8f;

namespace {
constexpr int T  = 2;
constexpr int N  = 10000;
constexpr int C  = 64;
constexpr int H  = 4;
constexpr int Co = 64;
constexpr int HC = H * Co;        // 256
constexpr int E  = 160000;
constexpr int M  = T * N;         // 20000 rows (T*N, B==1)
constexpr float NEG_SLOPE = 0.2f;
constexpr float LN_EPS    = 1e-5f;

// workspace layout in floats
constexpr size_t OFF_XL    = 0;
constexpr size_t OFF_XR    = OFF_XL    + (size_t)M * HC;      // 5.12M floats
constexpr size_t OFF_ALPHA = OFF_XR    + (size_t)M * HC;
constexpr size_t OFF_AMAX  = OFF_ALPHA + (size_t)T * E * H;   // uint-encoded
constexpr size_t OFF_DENOM = OFF_AMAX  + (size_t)T * N * H;
constexpr size_t OFF_AGG   = OFF_DENOM + (size_t)T * N * H;
constexpr size_t OFF_XSAM  = OFF_AGG   + (size_t)M * HC;
} // namespace

__device__ __forceinline__ v8f wmma_f32_4(v2f a, v2f b, v8f c) {
  // V_WMMA_F32_16X16X4_F32 : D = A(16x4) * B(4x16) + C, fp32 end to end
  return __builtin_amdgcn_wmma_f32_16x16x4_f32(
      /*neg_a=*/false, a, /*neg_b=*/false, b,
      /*c_mod=*/(short)0, c, /*reuse_a=*/false, /*reuse_b=*/false);
}

// order-preserving float<->uint encoding for atomicMax-based segment max
__device__ __forceinline__ unsigned f2ord(float f) {
  unsigned u = __float_as_uint(f);
  return (u & 0x80000000u) ? ~u : (u | 0x80000000u);
}
__device__ __forceinline__ float ord2f(unsigned u) {
  return __uint_as_float((u & 0x80000000u) ? (u & 0x7FFFFFFFu) : ~u);
}

__device__ __forceinline__ float wave_sum(float s) {
  #pragma unroll
  for (int off = 16; off > 0; off >>= 1) s += __shfl_xor(s, off, 32);
  return s;
}

// ---- 1) x @ W + b  (out: M x 256). One wave per 16x64 tile: the A fragment
//         is loaded once per k-step and reused across 4 WMMAs (4 accumulators).
__global__ __launch_bounds__(32)
void sam_gemm_lr(const float* __restrict__ x, const float* __restrict__ W,
                 const float* __restrict__ bv, float* __restrict__ out) {
  const int lane  = threadIdx.x;
  const int l15   = lane & 15;
  const int mrow  = (blockIdx.x << 4) + l15;   // A row for this lane
  const int nbase = (blockIdx.y << 6);         // 64-wide column stripe
  const int khalf = (lane >> 4) << 1;          // 0 or 2
  const float* __restrict__ arow = x + (size_t)mrow * C;
  v8f acc0 = {}, acc1 = {}, acc2 = {}, acc3 = {};
  #pragma unroll
  for (int k = 0; k < C; k += 4) {
    v2f a;
    a.x = arow[k + khalf];
    a.y = arow[k + khalf + 1];
    const float* __restrict__ w0 = W + (size_t)(k + khalf)     * HC + nbase + l15;
    const float* __restrict__ w1 = W + (size_t)(k + khalf + 1) * HC + nbase + l15;
    v2f b0 = {w0[0],  w1[0]};
    v2f b1 = {w0[16], w1[16]};
    v2f b2 = {w0[32], w1[32]};
    v2f b3 = {w0[48], w1[48]};
    acc0 = wmma_f32_4(a, b0, acc0);
    acc1 = wmma_f32_4(a, b1, acc1);
    acc2 = wmma_f32_4(a, b2, acc2);
    acc3 = wmma_f32_4(a, b3, acc3);
  }
  const int   mbase = (blockIdx.x << 4) + ((lane >> 4) << 3);
  const float bb0 = bv[nbase + l15];
  const float bb1 = bv[nbase + l15 + 16];
  const float bb2 = bv[nbase + l15 + 32];
  const float bb3 = bv[nbase + l15 + 48];
  #pragma unroll
  for (int r = 0; r < 8; ++r) {
    float* __restrict__ orow = out + (size_t)(mbase + r) * HC + nbase + l15;
    orow[0]  = acc0[r] + bb0;
    orow[16] = acc1[r] + bb1;
    orow[32] = acc2[r] + bb2;
    orow[48] = acc3[r] + bb3;
  }
}

// ---------------- 2) per-edge alpha + segment max (wave per (t,e,h)) -------
__global__ __launch_bounds__(256)
void sam_edge_alpha(const float* __restrict__ xl, const float* __restrict__ xr,
                    const int* __restrict__ ei, const float* __restrict__ att,
                    float* __restrict__ alpha, unsigned* __restrict__ amax) {
  const int gw   = (int)((blockIdx.x * 256u + threadIdx.x) >> 5);
  const int lane = threadIdx.x & 31;
  if (gw >= T * E * H) return;
  const int h = gw & (H - 1);
  const int r = gw >> 2;          // t*E + e
  const int e = r % E;
  const int t = r / E;
  const int src = ei[e];
  const int dst = ei[E + e];
  const float* __restrict__ pj = xl + (size_t)(t * N + src) * HC + h * Co;
  const float* __restrict__ pi = xr + (size_t)(t * N + dst) * HC + h * Co;
  const float* __restrict__ pa = att + h * Co;
  float s = 0.f;
  #pragma unroll
  for (int u = 0; u < 2; ++u) {
    const int co = lane + u * 32;
    float m = pi[co] + pj[co];
    m = (m > 0.f) ? m : m * NEG_SLOPE;
    s += m * pa[co];
  }
  s = wave_sum(s);
  if (lane == 0) {
    alpha[gw] = s;
    atomicMax(&amax[(size_t)(t * N + dst) * H + h], f2ord(s));
  }
}

// ---------------- 3) exp(alpha - max) + segment sum (thread per item) ------
__global__ __launch_bounds__(256)
void sam_edge_exp(const int* __restrict__ ei, const unsigned* __restrict__ amax,
                  float* __restrict__ denom, float* alpha /*in/out*/) {
  const int idx = (int)(blockIdx.x * 256u + threadIdx.x);
  if (idx >= T * E * H) return;
  const int h = idx & (H - 1);
  const int r = idx >> 2;
  const int e = r % E;
  const int t = r / E;
  const int dst = ei[E + e];
  const size_t seg = (size_t)(t * N + dst) * H + h;
  const float a = __expf(alpha[idx] - ord2f(amax[seg]));
  alpha[idx] = a;
  atomicAdd(&denom[seg], a);
}

// ---------------- 4) weighted scatter-add (wave per (t,e,h)) ---------------
__global__ __launch_bounds__(256)
void sam_edge_agg(const float* __restrict__ xl, const int* __restrict__ ei,
                  const float* __restrict__ alpha, const float* __restrict__ denom,
                  float* __restrict__ agg) {
  const int gw   = (int)((blockIdx.x * 256u + threadIdx.x) >> 5);
  const int lane = threadIdx.x & 31;
  if (gw >= T * E * H) return;
  const int h = gw & (H - 1);
  const int r = gw >> 2;
  const int e = r % E;
  const int t = r / E;
  const int src = ei[e];
  const int dst = ei[E + e];
  const float attn = alpha[gw] / denom[(size_t)(t * N + dst) * H + h];
  const float* __restrict__ pj = xl + (size_t)(t * N + src) * HC + h * Co;
  float* __restrict__ po = agg + (size_t)(t * N + dst) * HC + h * Co;
  #pragma unroll
  for (int u = 0; u < 2; ++u) {
    const int co = lane + u * 32;
    atomicAdd(&po[co], pj[co] * attn);
  }
}

// ---- 5) relu(agg+bias) @ W_proj + b_proj + x. One wave per 16x64 tile (all
//         64 output cols): the fused relu(+bias) A fragment is built once per
//         k-step and reused across 4 WMMAs.
__global__ __launch_bounds__(32)
void sam_gemm_proj(const float* __restrict__ agg, const float* __restrict__ bias,
                   const float* __restrict__ Wp, const float* __restrict__ bp,
                   const float* __restrict__ x, float* __restrict__ xsam) {
  const int lane  = threadIdx.x;
  const int l15   = lane & 15;
  const int mrow  = (blockIdx.x << 4) + l15;
  const int khalf = (lane >> 4) << 1;
  const float* __restrict__ arow = agg + (size_t)mrow * HC;
  v8f acc0 = {}, acc1 = {}, acc2 = {}, acc3 = {};
  #pragma unroll 8
  for (int k = 0; k < HC; k += 4) {
    v2f a;
    a.x = fmaxf(arow[k + khalf]     + bias[k + khalf],     0.f); // relu(out+bias)
    a.y = fmaxf(arow[k + khalf + 1] + bias[k + khalf + 1], 0.f);
    const float* __restrict__ w0 = Wp + (size_t)(k + khalf)     * Co + l15;
    const float* __restrict__ w1 = Wp + (size_t)(k + khalf + 1) * Co + l15;
    v2f b0 = {w0[0],  w1[0]};
    v2f b1 = {w0[16], w1[16]};
    v2f b2 = {w0[32], w1[32]};
    v2f b3 = {w0[48], w1[48]};
    acc0 = wmma_f32_4(a, b0, acc0);
    acc1 = wmma_f32_4(a, b1, acc1);
    acc2 = wmma_f32_4(a, b2, acc2);
    acc3 = wmma_f32_4(a, b3, acc3);
  }
  const int   mbase = (blockIdx.x << 4) + ((lane >> 4) << 3);
  const float bb0 = bp[l15];
  const float bb1 = bp[l15 + 16];
  const float bb2 = bp[l15 + 32];
  const float bb3 = bp[l15 + 48];
  #pragma unroll
  for (int r = 0; r < 8; ++r) {
    const size_t row = (size_t)(mbase + r);
    const float* __restrict__ xrow = x    + row * Co + l15;
    float*       __restrict__ orow = xsam + row * Co + l15;
    orow[0]  = acc0[r] + bb0 + xrow[0];   // residual
    orow[16] = acc1[r] + bb1 + xrow[16];
    orow[32] = acc2[r] + bb2 + xrow[32];
    orow[48] = acc3[r] + bb3 + xrow[48];
  }
}

// ---------------- 6) LayerNorm(64) + ReLU (wave per row) -------------------
__global__ __launch_bounds__(256)
void sam_ln_relu(const float* __restrict__ xsam, const float* __restrict__ gamma,
                 const float* __restrict__ beta, float* __restrict__ out) {
  const int row  = (int)((blockIdx.x * 256u + threadIdx.x) >> 5);
  const int lane = threadIdx.x & 31;
  if (row >= M) return;
  const float v0 = xsam[(size_t)row * Co + lane];
  const float v1 = xsam[(size_t)row * Co + lane + 32];
  const float s  = wave_sum(v0 + v1);
  const float q  = wave_sum(v0 * v0 + v1 * v1);
  const float mean = s * (1.f / Co);
  const float var  = q * (1.f / Co) - mean * mean;
  const float rstd = rsqrtf(var + LN_EPS);
  out[(size_t)row * Co + lane] =
      fmaxf((v0 - mean) * rstd * gamma[lane] + beta[lane], 0.f);
  out[(size_t)row * Co + lane + 32] =
      fmaxf((v1 - mean) * rstd * gamma[lane + 32] + beta[lane + 32], 0.f);
}

extern "C" void kernel_launch(void* const* d_in, const int* in_sizes, int n_in,
                              void* d_out, int out_size, void* d_ws, size_t ws_size,
                              hipStream_t stream) {
  (void)in_sizes; (void)n_in; (void)out_size; (void)ws_size;
  const float* x      = (const float*)d_in[0];
  const int*   ei     = (const int*)  d_in[1];
  const float* W_l    = (const float*)d_in[2];
  const float* b_l    = (const float*)d_in[3];
  const float* W_r    = (const float*)d_in[4];
  const float* b_r    = (const float*)d_in[5];
  const float* att    = (const float*)d_in[6];
  const float* bias   = (const float*)d_in[7];
  const float* W_proj = (const float*)d_in[8];
  const float* b_proj = (const float*)d_in[9];
  const float* gamma  = (const float*)d_in[10];
  const float* beta   = (const float*)d_in[11];
  float* out = (float*)d_out;
  float* ws  = (float*)d_ws;

  float*    xl    = ws + OFF_XL;
  float*    xr    = ws + OFF_XR;
  float*    alpha = ws + OFF_ALPHA;
  unsigned* amax  = (unsigned*)(ws + OFF_AMAX);
  float*    denom = ws + OFF_DENOM;
  float*    agg   = ws + OFF_AGG;
  float*    xsam  = ws + OFF_XSAM;

  // zero amax (ordered -inf == 0), denom, agg in one contiguous memset
  hipMemsetAsync(ws + OFF_AMAX, 0, (OFF_XSAM - OFF_AMAX) * sizeof(float), stream);

  // 1) x_l, x_r projections (WMMA fp32), one wave per 16x64 output tile
  dim3 gLR(M / 16, HC / 64);
  sam_gemm_lr<<<gLR, 32, 0, stream>>>(x, W_l, b_l, xl);
  sam_gemm_lr<<<gLR, 32, 0, stream>>>(x, W_r, b_r, xr);

  // edge phase: T*E*H wave-items
  const int items      = T * E * H;                       // 1,280,000
  const int waveBlocks = (items * 32 + 255) / 256;        // 160,000
  sam_edge_alpha<<<waveBlocks, 256, 0, stream>>>(xl, xr, ei, att, alpha, amax);
  sam_edge_exp<<<(items + 255) / 256, 256, 0, stream>>>(ei, amax, denom, alpha);
  sam_edge_agg<<<waveBlocks, 256, 0, stream>>>(xl, ei, alpha, denom, agg);

  // 5) projection GEMM fused with relu(+bias) and residual; 16x64 per wave
  sam_gemm_proj<<<M / 16, 32, 0, stream>>>(agg, bias, W_proj, b_proj, x, xsam);

  // 6) LayerNorm + ReLU, wave per 64-wide row
  sam_ln_relu<<<(M * 32 + 255) / 256, 256, 0, stream>>>(xsam, gamma, beta, out);
}